// VectorQuantizerEMA_44607530336801
// MI455X (gfx1250) — compile-verified
//
#include <hip/hip_runtime.h>
#include <hip/hip_bf16.h>
#include <math.h>
#include <stdint.h>

#define NUM_EMB 8192
#define CDIM    256
#define BATCH   16
#define HW      1024          // 32*32
#define NTOK    (BATCH*HW)    // 16384
#define COMMITC 0.25f

typedef float v2f __attribute__((ext_vector_type(2)));
typedef float v8f __attribute__((ext_vector_type(8)));
typedef unsigned int v4u __attribute__((ext_vector_type(4)));
typedef int v4i __attribute__((ext_vector_type(4)));
typedef int v8i __attribute__((ext_vector_type(8)));

// ---------------------------------------------------------------------------
// Workspace layout (elements):
//   int   idx[NTOK] | int counts[NUM_EMB] | float loss_sum | float enorm[NUM_EMB]
// ---------------------------------------------------------------------------

__global__ void k_enorm(const float* __restrict__ E, float* __restrict__ enorm) {
    int r = blockIdx.x * blockDim.x + threadIdx.x;
    if (r >= NUM_EMB) return;
    const float4* row = (const float4*)(E + (size_t)r * CDIM);
    float s = 0.f;
#pragma unroll 4
    for (int i = 0; i < CDIM / 4; ++i) {
        float4 v = row[i];
        s += v.x * v.x + v.y * v.y + v.z * v.z + v.w * v.w;
    }
    enorm[r] = s;
}

// ---------------------------------------------------------------------------
// Distance GEMM + argmin.
//   Block = 128 threads (4 waves) sharing one 64-row A tile in LDS (64KB).
//   A^T staged via the Tensor Data Mover (TDM): a 2D D# tile of 64 contiguous
//   f32 x 256 lines with stride HW, issued by wave 0, awaited with
//   s_wait_tensorcnt, then published to the other waves via barrier.
//   Each wave sweeps 2048 of the 8192 codes in 16-col tiles with
//   V_WMMA_F32_16X16X4_F32 (4 M-blocked accumulators), keeping a PER-LANE
//   running (min, argmin); the 16-lane butterfly runs once at the end, and
//   the 4 per-wave winners are combined through the retired A-tile LDS.
// ---------------------------------------------------------------------------
__launch_bounds__(128)
__global__ void k_argmin(const float* __restrict__ in,
                         const float* __restrict__ E,
                         const float* __restrict__ enorm,
                         int*         __restrict__ idx_out) {
    __shared__ float sAT[CDIM * 64];   // [k][i], 65536 B; reused for combine

    const int tid     = threadIdx.x;
    const int lane    = tid & 31;
    const int wid     = tid >> 5;
    const int rowBase = blockIdx.x * 64;   // never crosses a batch image
    const int b       = rowBase / HW;
    const int hw0     = rowBase % HW;

#if __has_builtin(__builtin_amdgcn_tensor_load_to_lds) && \
    __has_builtin(__builtin_amdgcn_s_wait_tensorcnt)
    if (wid == 0) {
        // D# descriptor: 2D tile, x = 64 f32 (contiguous), y = 256 lines,
        // line stride = HW elements.  TDM writes LDS rows contiguously ->
        // exactly the sAT[c*64 + i] layout.
        const uint64_t ga = (uint64_t)(uintptr_t)(in + (size_t)b * CDIM * HW + hw0);
        const uint32_t lds_base = (uint32_t)(uintptr_t)(void*)sAT; // LDS offset
        v4u g0;
        g0[0] = 1u;                                   // count=1 (valid user D#)
        g0[1] = lds_base;                             // lds_addr
        g0[2] = (uint32_t)ga;                         // global_addr[31:0]
        g0[3] = (uint32_t)((ga >> 32) & 0x01FFFFFFu)  // global_addr[56:32]
                | 0x80000000u;                        // type=2 ("image")
        v8i g1;
        g1[0] = (int)(2u << 16);      // workgroup_mask=0, data_size=4B
        g1[1] = (int)(64u << 16);     // tensor_dim0 = 64  (bits 79:48, low)
        g1[2] = (int)(256u << 16);    // tensor_dim1 = 256 (bits 111:80, low)
        g1[3] = (int)(64u << 16);     // tile_dim0 = 64    (bits 127:112)
        g1[4] = 256;                  // tile_dim1 = 256, tile_dim2 = 0
        g1[5] = 1024;                 // tensor_dim0_stride = HW (bits 207:160)
        g1[6] = 0;
        g1[7] = 0;                    // tensor_dim1_stride = 0 (unused, 2D)
        v4i z4 = {0, 0, 0, 0};        // groups 2/3 unused for 2D tensors
        v8i z8 = {0, 0, 0, 0, 0, 0, 0, 0};
        __builtin_amdgcn_tensor_load_to_lds(g0, g1, z4, z4, z8, 0);
        __builtin_amdgcn_s_wait_tensorcnt(0);
    }
    __syncthreads();
#else
    {
        const int gbase = b * (CDIM * HW / 4) + hw0 / 4;
        float4* s4 = (float4*)sAT;
        const float4* in4 = (const float4*)in;
        for (int f = tid; f < CDIM * 16; f += 128) {
            int c = f >> 4, i4 = f & 15;
            s4[c * 16 + i4] = in4[gbase + c * (HW / 4) + i4];
        }
    }
    __syncthreads();
#endif

    const int col16 = lane & 15;          // column slot / row within M-tile
    const int kb    = (lane >> 4) << 1;   // K sub-offset: 0 or 2

    float rbv[4][8];
    int   rbi[4][8];
#pragma unroll
    for (int mi = 0; mi < 4; ++mi)
#pragma unroll
        for (int j = 0; j < 8; ++j) { rbv[mi][j] = 3.4e38f; rbi[mi][j] = 0; }

    const int t0 = wid * (NUM_EMB / 16 / 4);      // 128 tiles per wave
    for (int t = t0; t < t0 + NUM_EMB / 16 / 4; ++t) {
        const int colBase = t * 16;
        v8f acc[4] = {};

        const float* Erow = E + (size_t)(colBase + col16) * CDIM + kb;
#pragma unroll 4
        for (int kk = 0; kk < CDIM / 4; ++kk) {
            const int c0 = kk * 4;
            v2f bv = *(const v2f*)(Erow + c0);
#pragma unroll
            for (int mi = 0; mi < 4; ++mi) {
                const int ib = mi * 16 + col16;
                v2f av;
                av.x = sAT[(c0 + kb) * 64 + ib];
                av.y = sAT[(c0 + kb + 1) * 64 + ib];
                acc[mi] = __builtin_amdgcn_wmma_f32_16x16x4_f32(
                    false, av, false, bv, (short)0, acc[mi], false, false);
            }
        }

        // per-lane running argmin (branchless; strict '<' + ascending tiles
        // preserves first-min semantics per lane)
        const float en = enorm[colBase + col16];
        const int   si = colBase + col16;
#pragma unroll
        for (int mi = 0; mi < 4; ++mi)
#pragma unroll
            for (int j = 0; j < 8; ++j) {
                float sv = fmaf(-2.0f, acc[mi][j], en);
                bool better = sv < rbv[mi][j];
                rbv[mi][j] = better ? sv : rbv[mi][j];
                rbi[mi][j] = better ? si : rbi[mi][j];
            }
    }

    // one 16-lane min+argmin butterfly per accumulator value
#pragma unroll
    for (int mi = 0; mi < 4; ++mi)
#pragma unroll
        for (int j = 0; j < 8; ++j) {
            float v = rbv[mi][j];
            int   ix = rbi[mi][j];
#pragma unroll
            for (int m = 1; m < 16; m <<= 1) {
                float ov = __shfl_xor(v, m, 32);
                int   oi = __shfl_xor(ix, m, 32);
                bool better = (ov < v) || (ov == v && oi < ix);
                v  = better ? ov : v;
                ix = better ? oi : ix;
            }
            rbv[mi][j] = v;
            rbi[mi][j] = ix;
        }

    // combine the 4 per-wave winners through the retired A-tile LDS
    __syncthreads();                       // everyone done reading sAT
    float* cbv = sAT;                      // [4][64]
    int*   cbi = (int*)(sAT + 4 * 64);     // [4][64]
    if (col16 == 0) {
        const int half = lane >> 4;        // 0 -> rows j, 1 -> rows j+8
#pragma unroll
        for (int mi = 0; mi < 4; ++mi)
#pragma unroll
            for (int j = 0; j < 8; ++j) {
                const int rl = mi * 16 + half * 8 + j;
                cbv[wid * 64 + rl] = rbv[mi][j];
                cbi[wid * 64 + rl] = rbi[mi][j];
            }
    }
    __syncthreads();
    if (tid < 64) {
        float bv = cbv[tid];
        int   bi = cbi[tid];
#pragma unroll
        for (int w = 1; w < 4; ++w) {
            float ov = cbv[w * 64 + tid];
            int   oi = cbi[w * 64 + tid];
            bool better = (ov < bv) || (ov == bv && oi < bi);
            bv = better ? ov : bv;
            bi = better ? oi : bi;
        }
        idx_out[rowBase + tid] = bi;
    }
}

// ---------------------------------------------------------------------------
__global__ void k_quant(const float* __restrict__ in,
                        const float* __restrict__ E,
                        const int*   __restrict__ idx,
                        float*       __restrict__ quant_out,
                        float*       __restrict__ loss_sum) {
    const int bc = blockIdx.x;
    const int b  = bc >> 8;
    const int c  = bc & 255;
    const size_t base = (size_t)b * (CDIM * HW) + (size_t)c * HW;
    const int* idxb = idx + b * HW;

    float part = 0.f;
    for (int hw = threadIdx.x; hw < HW; hw += blockDim.x) {
        float q = E[(size_t)idxb[hw] * CDIM + c];
        float x = in[base + hw];
        float d = q - x;
        part += d * d;
        quant_out[base + hw] = q;
    }
    __shared__ float red[256];
    red[threadIdx.x] = part;
    __syncthreads();
    for (int s = 128; s > 0; s >>= 1) {
        if (threadIdx.x < s) red[threadIdx.x] += red[threadIdx.x + s];
        __syncthreads();
    }
    if (threadIdx.x == 0) atomicAdd(loss_sum, red[0]);
}

__global__ void k_scatter(const int* __restrict__ idx,
                          float* __restrict__ enc,
                          int*   __restrict__ counts) {
    int n = blockIdx.x * blockDim.x + threadIdx.x;
    if (n < NTOK) {
        int k = idx[n];
        enc[(size_t)n * NUM_EMB + k] = 1.0f;
        atomicAdd(&counts[k], 1);
    }
}

__global__ void k_final(const int*   __restrict__ counts,
                        const float* __restrict__ loss_sum,
                        float* __restrict__ loss_out,
                        float* __restrict__ perp_out) {
    float s = 0.f;
    for (int k = threadIdx.x; k < NUM_EMB; k += blockDim.x) {
        float p = (float)counts[k] * (1.0f / (float)NTOK);
        s += p * logf(p + 1e-10f);
    }
    __shared__ float red[256];
    red[threadIdx.x] = s;
    __syncthreads();
    for (int st = 128; st > 0; st >>= 1) {
        if (threadIdx.x < st) red[threadIdx.x] += red[threadIdx.x + st];
        __syncthreads();
    }
    if (threadIdx.x == 0) {
        *perp_out = expf(-red[0]);
        *loss_out = COMMITC * (*loss_sum) / (float)(NTOK * CDIM);
    }
}

extern "C" void kernel_launch(void* const* d_in, const int* in_sizes, int n_in,
                              void* d_out, int out_size, void* d_ws, size_t ws_size,
                              hipStream_t stream) {
    const float* in = (const float*)d_in[0];   // [16,256,32,32] f32
    const float* E  = (const float*)d_in[1];   // [8192,256] f32

    // d_out: loss(1) | quantized(4194304) | perplexity(1) | encodings(134217728)
    float* out      = (float*)d_out;
    float* loss_o   = out;
    float* quant_o  = out + 1;
    float* perp_o   = out + 1 + (size_t)BATCH * CDIM * HW;
    float* enc_o    = out + 2 + (size_t)BATCH * CDIM * HW;

    int*   idx      = (int*)d_ws;
    int*   counts   = idx + NTOK;
    float* loss_sum = (float*)(counts + NUM_EMB);
    float* enorm    = loss_sum + 1;

    (void)hipMemsetAsync(enc_o, 0, (size_t)NTOK * NUM_EMB * sizeof(float), stream);
    (void)hipMemsetAsync(counts, 0, (NUM_EMB + 1) * sizeof(int), stream);

    k_enorm<<<NUM_EMB / 256, 256, 0, stream>>>(E, enorm);
    k_argmin<<<NTOK / 64, 128, 0, stream>>>(in, E, enorm, idx);
    k_quant<<<BATCH * CDIM, 256, 0, stream>>>(in, E, idx, quant_o, loss_sum);
    k_scatter<<<NTOK / 256, 256, 0, stream>>>(idx, enc_o, counts);
    k_final<<<1, 256, 0, stream>>>(counts, loss_sum, loss_o, perp_o);
}